// NMS_20710332301630
// MI455X (gfx1250) — compile-verified
//
#include <hip/hip_runtime.h>
#include <cstdint>
#include <math.h>

#define N_ANCH    172032
#define MAX_OUT   200
#define NMS_T     0.4f
#define VAR0      0.1f
#define VAR1      0.2f
#define NMS_BLOCK 1024
#define NBLK      16     // persistent blocks (WGPs) cooperating on NMS

// ---------------- CDNA5 async-to-LDS feature detection -----------------
#if defined(__AMDGCN__) && __has_builtin(__builtin_amdgcn_global_load_async_to_lds_b128)
#define USE_ASYNC 1
#else
#define USE_ASYNC 0
#endif

#if defined(__AMDGCN__) && __has_builtin(__builtin_amdgcn_s_wait_asynccnt)
#define WAIT_ASYNC(n) __builtin_amdgcn_s_wait_asynccnt(n)
#elif USE_ASYNC
#define WAIT_ASYNC(n) asm volatile("s_wait_asynccnt %0" :: "n"(n) : "memory")
#else
#define WAIT_ASYNC(n) do {} while (0)
#endif

#if USE_ASYNC
typedef int v4i __attribute__((vector_size(4 * sizeof(int))));
typedef __attribute__((address_space(1))) v4i GV;  // global
typedef __attribute__((address_space(3))) v4i LV;  // LDS
__device__ __forceinline__ void async_copy16(const void* g, void* l) {
  // global -> LDS, 16B per active lane; 1 ASYNCcnt tick per wave-instruction
  __builtin_amdgcn_global_load_async_to_lds_b128(
      (GV*)(unsigned long long)(uintptr_t)g,
      (LV*)(unsigned int)(uintptr_t)l,
      0, 0);
}
#endif

__device__ __forceinline__ void backoff() {
#if defined(__AMDGCN__) && __has_builtin(__builtin_amdgcn_s_sleep)
  __builtin_amdgcn_s_sleep(1);
#endif
}

// ---------------------------- kernels ----------------------------------

__global__ void init_kernel(int* cnt, unsigned* barctr) { *cnt = 0; *barctr = 0u; }

// Decode + threshold-compact candidates. ~22MB traffic, HBM-bound (~1us).
// key = (float_bits(score) << 32) | ~orig_idx : u64 max == (score desc, idx asc);
// key==0 means "suppressed/dead" (live scores > 0.4 => high word >= 0x3ECCCCCD).
__global__ void decode_kernel(const float* __restrict__ cls,
                              const float* __restrict__ reg,
                              const float* __restrict__ lnd,
                              const float* __restrict__ anc,
                              int*    __restrict__ cnt,
                              float4* __restrict__ cbox,
                              unsigned long long* __restrict__ ckey,
                              float*  __restrict__ clnd) {
  int i = blockIdx.x * blockDim.x + threadIdx.x;
  if (i >= N_ANCH) return;
  float sc = cls[2 * i + 1];
  if (!(sc > NMS_T)) return;  // can never be selected: s0 = -inf forever

  float4 a = ((const float4*)anc)[i];   // x_a, y_a, w_a, h_a
  float4 r = ((const float4*)reg)[i];
  float dx = r.x * VAR0, dy = r.y * VAR0, dw = r.z * VAR1, dh = r.w * VAR1;
  float xc = dx * a.z + a.x;
  float yc = dy * a.w + a.y;
  float w  = expf(dw) * a.z;
  float h  = expf(dh) * a.w;
  float4 b;
  b.x = yc - 0.5f * h;  // y1
  b.y = xc - 0.5f * w;  // x1
  b.z = yc + 0.5f * h;  // y2
  b.w = xc + 0.5f * w;  // x2

  int pos = atomicAdd(cnt, 1);
  cbox[pos] = b;
  ckey[pos] = ((unsigned long long)__float_as_uint(sc) << 32) | (unsigned)(~(unsigned)i);

  const float2* lp = (const float2*)(lnd + 10 * i);
  float* lo = clnd + 10 * pos;
#pragma unroll
  for (int k = 0; k < 5; ++k) {
    float2 v = lp[k];
    lo[2 * k]     = v.x * VAR0 * a.z + a.x;
    lo[2 * k + 1] = v.y * VAR0 * a.w + a.y;
  }
}

__device__ __forceinline__ void kred(unsigned long long& k, int& c) {
#pragma unroll
  for (int o = 16; o > 0; o >>= 1) {  // wave32 butterfly max on u64 key
    unsigned long long ok = __shfl_xor(k, o);
    int oc = __shfl_xor(c, o);
    if (ok > k) { k = ok; c = oc; }
  }
}

__device__ __forceinline__ void block_red(unsigned long long& k, int& c,
                                          unsigned long long* s_k, int* s_c, int tid) {
  kred(k, c);
  if ((tid & 31) == 0) { s_k[tid >> 5] = k; s_c[tid >> 5] = c; }
  __syncthreads();
  if (tid < 32) { k = s_k[tid]; c = s_c[tid]; kred(k, c); }  // thread 0 holds result
}

// One grid barrier per NMS iteration: monotonic counter, leader-spin.
__device__ __forceinline__ void grid_barrier(unsigned* ctr, unsigned target) {
  __threadfence();      // release this block's global writes (partials, keys)
  __syncthreads();
  if (threadIdx.x == 0) {
#if __has_builtin(__hip_atomic_fetch_add) && __has_builtin(__hip_atomic_load)
    __hip_atomic_fetch_add(ctr, 1u, __ATOMIC_ACQ_REL, __HIP_MEMORY_SCOPE_AGENT);
    while (__hip_atomic_load(ctr, __ATOMIC_RELAXED, __HIP_MEMORY_SCOPE_AGENT) < target) backoff();
#else
    atomicAdd(ctr, 1u);
    while (*(volatile unsigned*)ctr < target) backoff();
#endif
  }
  __syncthreads();
  __threadfence();      // acquire on every thread (both CUs of the WGP)
}

// Persistent cooperative NMS across NBLK WGPs.
// Iteration t (t>=1) fuses "suppress vs best_{t-1}" with "argmax for best_t":
// one pass over L2-resident candidates per selected output.
__global__ void __launch_bounds__(NMS_BLOCK)
nms_kernel(const int* __restrict__ cnt,
           const float4* __restrict__ cbox,
           unsigned long long* __restrict__ ckey,
           int* __restrict__ selidx,
           int* __restrict__ selval,
           unsigned long long* __restrict__ pkey,   // [2][NBLK] partial keys
           int* __restrict__ pidx,                  // [2][NBLK] partial idxs
           unsigned* __restrict__ barctr) {
  __shared__ unsigned long long s_k[32];
  __shared__ int   s_c[32];
  __shared__ unsigned long long sh_bkey;
  __shared__ int   sh_bidx;
  __shared__ float4 sh_bbox;
#if USE_ASYNC
  __shared__ float4 s_tiles[2][NMS_BLOCK];  // 32 KB double buffer for box tiles
#endif

  const int tid = threadIdx.x;
  const int b = blockIdx.x;
  const int M = *cnt;
  const int tilesTotal = (M + NMS_BLOCK - 1) / NMS_BLOCK;  // 1024-wide tiles, tile t owned by block t%NBLK
  unsigned epoch = 0;

  // ---------------- pass 0: pure argmax over initial keys ----------------
  {
    unsigned long long k = 0; int c = -1;
    for (int t = b; t < tilesTotal; t += NBLK) {
      int j = t * NMS_BLOCK + tid;
      __builtin_prefetch((const void*)(ckey + j + NBLK * NMS_BLOCK), 0, 0);
      if (j < M) {
        unsigned long long kk = ckey[j];
        if (kk > k) { k = kk; c = j; }
      }
    }
    block_red(k, c, s_k, s_c, tid);
    if (tid == 0) { pkey[b] = k; pidx[b] = c; }   // parity-0 buffer
    grid_barrier(barctr, (++epoch) * NBLK);
    // every block reduces the NBLK partials (identical, deterministic result)
    unsigned long long rk = 0; int rc = -1;
    if (tid < NBLK) { rk = pkey[tid]; rc = pidx[tid]; }
    if (tid < 32) {
      kred(rk, rc);
      if (tid == 0) {
        sh_bkey = rk; sh_bidx = rc;
        if (rk != 0ULL) sh_bbox = cbox[rc];
      }
    }
    __syncthreads();
    if (b == 0 && tid == 0) {
      bool v = sh_bkey != 0ULL;
      selidx[0] = v ? sh_bidx : -1;
      selval[0] = v ? 1 : 0;
    }
  }

  // ---------------- iterations 1..199: fused suppress+argmax ----------------
  int it;
  for (it = 1; it < MAX_OUT; ++it) {
    if (sh_bkey == 0ULL) break;  // candidate set exhausted: same on all blocks

    const float by1 = sh_bbox.x, bxx1 = sh_bbox.y, by2 = sh_bbox.z, bxx2 = sh_bbox.w;
    const float ba = (by2 - by1) * (bxx2 - bxx1);  // == reference areas[i]

    unsigned long long k = 0; int c = -1;
#if USE_ASYNC
    {  // prologue: stage first owned tile
      int j0 = b * NMS_BLOCK + tid;
      if (b < tilesTotal && j0 < M) async_copy16(&cbox[j0], &s_tiles[0][tid]);
    }
    int bufIdx = 0;
    for (int t = b; t < tilesTotal; t += NBLK, bufIdx ^= 1) {
      int tn = t + NBLK;
      int waveBase = tid & ~31;
      bool issuesNext = (tn < tilesTotal) && (tn * NMS_BLOCK + waveBase < M);  // per-wave accounting
      if (issuesNext) {
        int jn = tn * NMS_BLOCK + tid;
        if (jn < M) async_copy16(&cbox[jn], &s_tiles[bufIdx ^ 1][tid]);
        WAIT_ASYNC(1);   // current tile landed; next still in flight
      } else {
        WAIT_ASYNC(0);
      }
      asm volatile("" ::: "memory");
      int j = t * NMS_BLOCK + tid;
      if (j < M) {
        unsigned long long kk = ckey[j];
        if (kk != 0ULL) {
          float4 bx = s_tiles[bufIdx][tid];
          float aj = (bx.z - bx.x) * (bx.w - bx.y);
          float ih = fmaxf(fminf(bx.z, by2) - fmaxf(bx.x, by1), 0.0f);
          float iw = fmaxf(fminf(bx.w, bxx2) - fmaxf(bx.y, bxx1), 0.0f);
          float inter = ih * iw;
          float iou = inter / (aj + ba - inter + 1e-12f);
          if (iou > NMS_T) ckey[j] = 0ULL;           // suppress (incl. best itself)
          else if (kk > k) { k = kk; c = j; }        // survivor competes for next best
        }
      }
    }
#else
    for (int t = b; t < tilesTotal; t += NBLK) {
      int j = t * NMS_BLOCK + tid;
      if (j < M) {
        unsigned long long kk = ckey[j];
        if (kk != 0ULL) {
          float4 bx = cbox[j];
          float aj = (bx.z - bx.x) * (bx.w - bx.y);
          float ih = fmaxf(fminf(bx.z, by2) - fmaxf(bx.x, by1), 0.0f);
          float iw = fmaxf(fminf(bx.w, bxx2) - fmaxf(bx.y, bxx1), 0.0f);
          float inter = ih * iw;
          float iou = inter / (aj + ba - inter + 1e-12f);
          if (iou > NMS_T) ckey[j] = 0ULL;
          else if (kk > k) { k = kk; c = j; }
        }
      }
    }
#endif
    block_red(k, c, s_k, s_c, tid);
    unsigned long long* pk = pkey + (it & 1) * NBLK;  // parity buffer: slow readers vs fast writers
    int* pc = pidx + (it & 1) * NBLK;
    if (tid == 0) { pk[b] = k; pc[b] = c; }
    grid_barrier(barctr, (++epoch) * NBLK);
    unsigned long long rk = 0; int rc = -1;
    if (tid < NBLK) { rk = pk[tid]; rc = pc[tid]; }
    if (tid < 32) {
      kred(rk, rc);
      if (tid == 0) {
        sh_bkey = rk; sh_bidx = rc;
        if (rk != 0ULL) sh_bbox = cbox[rc];
      }
    }
    __syncthreads();
    if (b == 0 && tid == 0) {
      bool v = sh_bkey != 0ULL;
      selidx[it] = v ? sh_bidx : -1;
      selval[it] = v ? 1 : 0;
    }
  }

  // fill any remaining (all-suppressed) slots; `it` is identical on all blocks
  if (b == 0) {
    for (int r = it + tid; r < MAX_OUT; r += NMS_BLOCK) { selidx[r] = -1; selval[r] = 0; }
  }
}

// Gather 200 selected boxes+landmarks, zeroing invalid slots.
__global__ void gather_kernel(const int* __restrict__ selidx,
                              const int* __restrict__ selval,
                              const float4* __restrict__ cbox,
                              const float* __restrict__ clnd,
                              float* __restrict__ out) {
  int t = blockIdx.x * blockDim.x + threadIdx.x;
  const int nbox = MAX_OUT * 4;
  const int ntot = MAX_OUT * 14;
  if (t >= ntot) return;
  if (t < nbox) {
    int slot = t >> 2, kk = t & 3;
    int c = selidx[slot];
    float v = 0.0f;
    if (selval[slot] && c >= 0) v = ((const float*)(cbox + c))[kk];
    out[t] = v;
  } else {
    int u = t - nbox;
    int slot = u / 10, kk = u % 10;
    int c = selidx[slot];
    float v = 0.0f;
    if (selval[slot] && c >= 0) v = clnd[c * 10 + kk];
    out[t] = v;
  }
}

// ---------------------------- launcher ----------------------------------
extern "C" void kernel_launch(void* const* d_in, const int* in_sizes, int n_in,
                              void* d_out, int out_size, void* d_ws, size_t ws_size,
                              hipStream_t stream) {
  const float* cls = (const float*)d_in[0];
  const float* reg = (const float*)d_in[1];
  const float* lnd = (const float*)d_in[2];
  const float* anc = (const float*)d_in[3];

  char* base = (char*)d_ws;
  size_t off = 0;
  int*      cnt    = (int*)(base + off);
  unsigned* barctr = (unsigned*)(base + off + 4);       off += 16;
  float4*   cbox   = (float4*)(base + off);             off += (size_t)N_ANCH * 16;
  unsigned long long* ckey = (unsigned long long*)(base + off); off += (size_t)N_ANCH * 8;
  float*    clnd   = (float*)(base + off);              off += (size_t)N_ANCH * 40;
  int*      selidx = (int*)(base + off);                off += 1024;
  int*      selval = (int*)(base + off);                off += 1024;
  unsigned long long* pkey = (unsigned long long*)(base + off); off += 2 * NBLK * 8;
  int*      pidx   = (int*)(base + off);                off += 2 * NBLK * 4;
  (void)off; (void)ws_size; (void)in_sizes; (void)n_in; (void)out_size;

  hipLaunchKernelGGL(init_kernel, dim3(1), dim3(1), 0, stream, cnt, barctr);
  hipLaunchKernelGGL(decode_kernel, dim3(N_ANCH / 256), dim3(256), 0, stream,
                     cls, reg, lnd, anc, cnt, cbox, ckey, clnd);
  hipLaunchKernelGGL(nms_kernel, dim3(NBLK), dim3(NMS_BLOCK), 0, stream,
                     cnt, cbox, ckey, selidx, selval, pkey, pidx, barctr);
  hipLaunchKernelGGL(gather_kernel, dim3((MAX_OUT * 14 + 255) / 256), dim3(256), 0, stream,
                     selidx, selval, cbox, clnd, (float*)d_out);
}